// KPlanesEmbedder_35373350650662
// MI455X (gfx1250) — compile-verified
//
#include <hip/hip_runtime.h>

#define FPL 8

typedef __attribute__((ext_vector_type(4))) float v4f;

struct Tables {
    const float* sp[4];   // spatial planes, per level: (3, H, W, 8) if TR else (3, 8, H, W)
    const float* tp[4];   // temporal planes, per level: (3, H, 300, 8) if TR else (3, 8, H, 300)
};

// ---------------------------------------------------------------------------
// Bilinear tap accumulation. All offsets are 32-bit (largest table = 25MB) so
// the backend can use saddr(base SGPR) + 32-bit voffset global loads.
// TR = true : plane layout (H, W, 8)  -> two 16B loads per corner
// TR = false: plane layout (8, H, W) -> strided scalar loads per corner
// ---------------------------------------------------------------------------
template <bool TR, bool ALIGN>
__device__ __forceinline__ void bilerp_acc(const float* __restrict__ plane,
                                           int W, int H, float gx, float gy,
                                           float acc[FPL])
{
    float ix, iy;
    if (ALIGN) {   // align_corners = True
        ix = (gx + 1.0f) * 0.5f * (float)(W - 1);
        iy = (gy + 1.0f) * 0.5f * (float)(H - 1);
    } else {       // align_corners = False
        ix = ((gx + 1.0f) * (float)W - 1.0f) * 0.5f;
        iy = ((gy + 1.0f) * (float)H - 1.0f) * 0.5f;
    }
    float x0f = floorf(ix), y0f = floorf(iy);
    float wx = ix - x0f, wy = iy - y0f;
    int x0i = (int)x0f, y0i = (int)y0f;
    unsigned x0 = (unsigned)min(max(x0i,     0), W - 1);
    unsigned x1 = (unsigned)min(max(x0i + 1, 0), W - 1);
    unsigned y0 = (unsigned)min(max(y0i,     0), H - 1);
    unsigned y1 = (unsigned)min(max(y0i + 1, 0), H - 1);
    float w00 = (1.0f - wx) * (1.0f - wy);
    float w01 = wx * (1.0f - wy);
    float w10 = (1.0f - wx) * wy;
    float w11 = wx * wy;

    if (TR) {
        unsigned i00 = (y0 * (unsigned)W + x0) * FPL;
        unsigned i01 = (y0 * (unsigned)W + x1) * FPL;
        unsigned i10 = (y1 * (unsigned)W + x0) * FPL;
        unsigned i11 = (y1 * (unsigned)W + x1) * FPL;
        v4f a0 = *(const v4f*)(plane + i00);
        v4f a1 = *(const v4f*)(plane + i00 + 4);
        v4f b0 = *(const v4f*)(plane + i01);
        v4f b1 = *(const v4f*)(plane + i01 + 4);
        v4f c0 = *(const v4f*)(plane + i10);
        v4f c1 = *(const v4f*)(plane + i10 + 4);
        v4f d0 = *(const v4f*)(plane + i11);
        v4f d1 = *(const v4f*)(plane + i11 + 4);
        #pragma unroll
        for (int k = 0; k < 4; ++k) {
            acc[k]     = fmaf(w00, a0[k], fmaf(w01, b0[k], fmaf(w10, c0[k], fmaf(w11, d0[k], acc[k]))));
            acc[k + 4] = fmaf(w00, a1[k], fmaf(w01, b1[k], fmaf(w10, c1[k], fmaf(w11, d1[k], acc[k + 4]))));
        }
    } else {
        unsigned HW  = (unsigned)H * (unsigned)W;
        unsigned b00 = y0 * (unsigned)W + x0;
        unsigned b01 = y0 * (unsigned)W + x1;
        unsigned b10 = y1 * (unsigned)W + x0;
        unsigned b11 = y1 * (unsigned)W + x1;
        #pragma unroll
        for (int f = 0; f < FPL; ++f) {
            unsigned o = (unsigned)f * HW;
            acc[f] = fmaf(w00, plane[b00 + o], fmaf(w01, plane[b01 + o],
                     fmaf(w10, plane[b10 + o], fmaf(w11, plane[b11 + o], acc[f]))));
        }
    }
}

// ---------------------------------------------------------------------------
// Layout transform: (3, 8, H, W) -> (3, H, W, 8).
// blockIdx.y = plane index (no integer divide). One thread per pixel:
// 8 coalesced strided reads, one contiguous 32B write.
// ---------------------------------------------------------------------------
__global__ __launch_bounds__(256) void kplanes_transpose(const float* __restrict__ src,
                                                         float* __restrict__ dst,
                                                         int HW)
{
    unsigned p = blockIdx.y;
    const float* s = src + (size_t)p * FPL * HW;
    float*       d = dst + (size_t)p * FPL * HW;
    int stride = gridDim.x * blockDim.x;
    for (int pix = blockIdx.x * blockDim.x + threadIdx.x; pix < HW; pix += stride) {
        float v[FPL];
        #pragma unroll
        for (int f = 0; f < FPL; ++f) v[f] = s[(unsigned)f * (unsigned)HW + (unsigned)pix];
        v4f* dv = (v4f*)(d + (unsigned)pix * FPL);
        dv[0] = (v4f){v[0], v[1], v[2], v[3]};
        dv[1] = (v4f){v[4], v[5], v[6], v[7]};
    }
}

// ---------------------------------------------------------------------------
// Main embed kernel. Grid-stride (2 points/thread) so we can prefetch the
// next iteration's streamed inputs (global_prefetch_b8). NT stores keep the
// 128MB output stream from evicting the ~61MB table set out of the 192MB L2.
// ---------------------------------------------------------------------------
template <bool TR>
__global__ __launch_bounds__(256) void kplanes_embed(const float* __restrict__ xyz,
                                                     const float* __restrict__ t,
                                                     Tables tb,
                                                     float* __restrict__ out,
                                                     int npts)
{
    const int sa[3] = {0, 0, 1};
    const int sb[3] = {1, 2, 2};
    int stride = gridDim.x * blockDim.x;
    for (int n = blockIdx.x * blockDim.x + threadIdx.x; n < npts; n += stride) {
        int nn = n + stride;
        if (nn < npts) {
            __builtin_prefetch(xyz + (size_t)3 * nn, 0, 0);   // global_prefetch_b8
            __builtin_prefetch(t + nn, 0, 0);
        }
        float c[3];
        c[0] = __builtin_nontemporal_load(xyz + (size_t)3 * n + 0);
        c[1] = __builtin_nontemporal_load(xyz + (size_t)3 * n + 1);
        c[2] = __builtin_nontemporal_load(xyz + (size_t)3 * n + 2);
        float tt  = __builtin_nontemporal_load(t + n);
        float gxT = 2.0f * tt - 1.0f;   // temporal-plane x coordinate (width=300)

        #pragma unroll
        for (int lvl = 0; lvl < 4; ++lvl) {
            const int r = 64 << lvl;
            float acc[FPL] = {0.f, 0.f, 0.f, 0.f, 0.f, 0.f, 0.f, 0.f};

            const float* sp = tb.sp[lvl];
            #pragma unroll
            for (int p = 0; p < 3; ++p)
                bilerp_acc<TR, false>(sp + (unsigned)p * (unsigned)(FPL * r * r),
                                      r, r, c[sa[p]], c[sb[p]], acc);

            const float* tp = tb.tp[lvl];
            #pragma unroll
            for (int p = 0; p < 3; ++p)
                bilerp_acc<TR, true>(tp + (unsigned)p * (unsigned)(FPL * r * 300),
                                     300, r, gxT, c[p], acc);

            float* orow = out + (size_t)n * 32 + lvl * FPL;
            __builtin_nontemporal_store((v4f){acc[0], acc[1], acc[2], acc[3]}, (v4f*)orow);
            __builtin_nontemporal_store((v4f){acc[4], acc[5], acc[6], acc[7]}, (v4f*)(orow + 4));
        }
    }
}

extern "C" void kernel_launch(void* const* d_in, const int* in_sizes, int n_in,
                              void* d_out, int out_size, void* d_ws, size_t ws_size,
                              hipStream_t stream)
{
    const float* xyz = (const float*)d_in[0];
    const float* t   = (const float*)d_in[1];
    // setup_inputs dict order interleaves: spatial_emb_i, temporal_emb_i
    const float* sp_in[4];
    const float* tp_in[4];
    for (int i = 0; i < 4; ++i) {
        sp_in[i] = (const float*)d_in[2 + 2 * i];
        tp_in[i] = (const float*)d_in[3 + 2 * i];
    }
    int npts = in_sizes[0] / 3;

    // Workspace layout for transposed tables
    size_t off = 0, spOff[4], tpOff[4];
    for (int i = 0; i < 4; ++i) {
        size_t r = (size_t)64 << i;
        spOff[i] = off; off += 3 * FPL * r * r;
        tpOff[i] = off; off += 3 * FPL * r * 300;
    }
    size_t needBytes = off * sizeof(float);

    int mainThreads = (npts + 1) / 2;              // 2 points per thread (prefetch distance)
    int mainBlocks  = (mainThreads + 255) / 256;

    Tables tb;
    if (ws_size >= needBytes) {
        float* ws = (float*)d_ws;
        for (int i = 0; i < 4; ++i) {
            int r = 64 << i;
            int HWs = r * r;
            int HWt = r * 300;
            dim3 gs((HWs + 255) / 256, 3);
            dim3 gt((HWt + 255) / 256, 3);
            kplanes_transpose<<<gs, 256, 0, stream>>>(sp_in[i], ws + spOff[i], HWs);
            kplanes_transpose<<<gt, 256, 0, stream>>>(tp_in[i], ws + tpOff[i], HWt);
            tb.sp[i] = ws + spOff[i];
            tb.tp[i] = ws + tpOff[i];
        }
        kplanes_embed<true><<<mainBlocks, 256, 0, stream>>>(xyz, t, tb, (float*)d_out, npts);
    } else {
        for (int i = 0; i < 4; ++i) { tb.sp[i] = sp_in[i]; tb.tp[i] = tp_in[i]; }
        kplanes_embed<false><<<mainBlocks, 256, 0, stream>>>(xyz, t, tb, (float*)d_out, npts);
    }
}